// JBF_block_14379550507247
// MI455X (gfx1250) — compile-verified
//
#include <hip/hip_runtime.h>

typedef __attribute__((ext_vector_type(2))) float v2f;
typedef __attribute__((ext_vector_type(8))) float v8f;

#define PATCH 125          // 5^3 patch elements per voxel
#define KPAD  128          // GEMM K padded to multiple of 4
#define KPITCH 132         // LDS row pitch: 132 % 64 == 4 -> conflict-free b64 frags
#define NOUT  27           // 3^3 conv outputs per voxel
#define NPAD  32           // GEMM N padded to 2x16
#define HW    128
#define DEPTH 18
#define DOUT  16

__global__ __launch_bounds__(128) void jbf_wmma_kernel(
    const float* __restrict__ x,
    const float* __restrict__ dn,
    const float* __restrict__ gi,
    const float* __restrict__ dw,
    const float* __restrict__ dbp,
    const float* __restrict__ rw,
    const float* __restrict__ rbp,
    float* __restrict__ out)
{
    // Per-wave staged A-matrices (coalesced fill, conflict-free fragment reads).
    __shared__ float sA[4][2][16][KPITCH];
    // Transposed sparse-conv-as-GEMM weight matrices: sW[n][k], zero padded.
    __shared__ float sWd[NPAD][KPITCH];
    __shared__ float sWr[NPAD][KPITCH];
    __shared__ float sWgt[4][16][NPAD];   // per-wave bilateral weights [voxel][o]
    __shared__ float s_dw[NOUT];
    __shared__ float s_rw[NOUT];

    const int t    = threadIdx.x;
    const int wave = t >> 5;
    const int lane = t & 31;

    if (t < NOUT) { s_dw[t] = dw[t]; s_rw[t] = rw[t]; }
    __syncthreads();

    const long long m0 = (long long)blockIdx.x * 64 + wave * 16;

    // ---- Build 128x32 GEMM B-matrices from the 27 conv taps (both convs) ----
    {
        const int n  = t >> 2;                 // 0..31 (output index o, padded)
        const int ks = (t & 3) * 32;           // k chunk start
        const int oz = n / 9, oy = (n - oz * 9) / 3, ox = n % 3;
        for (int kk = 0; kk < 32; ++kk) {
            const int k = ks + kk;
            float vd = 0.f, vr = 0.f;
            if (n < NOUT && k < PATCH) {
                const int pz = k / 25, rem = k - pz * 25;
                const int py = rem / 5, px = rem - py * 5;
                const int wz = pz - oz, wy = py - oy, wx = px - ox;
                if ((unsigned)wz < 3u && (unsigned)wy < 3u && (unsigned)wx < 3u) {
                    const int wi = (wz * 3 + wy) * 3 + wx;
                    vd = s_dw[wi];
                    vr = s_rw[wi];
                }
            }
            sWd[n][k] = vd;
            sWr[n][k] = vr;
        }
    }

    // ---- Stage 16 patch rows x 2 tensors into LDS, coalesced (128B/instr) ----
    // col >= 125 written as 0 => zero-padded K tail, no uninitialized LDS.
    for (int rt = 0; rt < 32; ++rt) {
        const int r    = rt & 15;
        const int tsel = rt >> 4;
        const float* grow = (tsel ? gi : dn) + (m0 + r) * PATCH;
#pragma unroll
        for (int j = 0; j < 4; ++j) {
            const int col = lane + 32 * j;     // 0..127
            float v = 0.f;
            if (col < PATCH) v = grow[col];
            sA[wave][tsel][r][col] = v;
        }
    }
    __syncthreads();

    // ---- GEMM: [16 voxels x 128] x [128 x 32], fp32 WMMA, two tensors ----
    const int arow  = lane & 15;               // A-matrix M row (voxel in tile)
    const int khalf = (lane >> 4) << 1;        // A/B fragment K sub-offset: 0 or 2
    const int ncol  = lane & 15;               // B/C N column

    v8f cd0 = {0,0,0,0,0,0,0,0};
    v8f cd1 = cd0, cr0 = cd0, cr1 = cd0;

    for (int kc = 0; kc < KPAD / 4; ++kc) {
        const int k0 = kc * 4 + khalf;         // even -> 8B-aligned b64 DS loads
        v2f ad  = { sA[wave][0][arow][k0], sA[wave][0][arow][k0 + 1] };
        v2f ar  = { sA[wave][1][arow][k0], sA[wave][1][arow][k0 + 1] };
        v2f bd0 = { sWd[ncol][k0],      sWd[ncol][k0 + 1] };
        v2f bd1 = { sWd[ncol + 16][k0], sWd[ncol + 16][k0 + 1] };
        v2f br0 = { sWr[ncol][k0],      sWr[ncol][k0 + 1] };
        v2f br1 = { sWr[ncol + 16][k0], sWr[ncol + 16][k0 + 1] };
        cd0 = __builtin_amdgcn_wmma_f32_16x16x4_f32(false, ad, false, bd0, (short)0, cd0, false, false);
        cd1 = __builtin_amdgcn_wmma_f32_16x16x4_f32(false, ad, false, bd1, (short)0, cd1, false, false);
        cr0 = __builtin_amdgcn_wmma_f32_16x16x4_f32(false, ar, false, br0, (short)0, cr0, false, false);
        cr1 = __builtin_amdgcn_wmma_f32_16x16x4_f32(false, ar, false, br1, (short)0, cr1, false, false);
    }

    // ---- bias + ReLU + product -> bilateral weights staged in LDS ----
    const float db = dbp[0];
    const float rb = rbp[0];
    {
        const int mbase = (lane >> 4) << 3;    // C layout: lanes 16-31 hold M=8..15
#pragma unroll
        for (int v = 0; v < 8; ++v) {
            const int m = mbase + v;
            float w0 = fmaxf(cd0[v] + db, 0.f) * fmaxf(cr0[v] + rb, 0.f) + 1e-10f;
            float w1 = fmaxf(cd1[v] + db, 0.f) * fmaxf(cr1[v] + rb, 0.f) + 1e-10f;
            sWgt[wave][m][ncol]      = w0;
            sWgt[wave][m][ncol + 16] = w1;     // o>=27 written but never read
        }
    }
    __syncthreads();

    // ---- weighted local sum over x neighborhood; 2 lanes per voxel ----
    {
        const int m = lane & 15;
        const long long gm = m0 + m;           // flat (b, dz, y, xw) voxel index
        const int xw = (int)( gm        & (HW - 1));
        const int y  = (int)((gm >> 7)  & (HW - 1));
        const int dz = (int)((gm >> 14) & (DOUT - 1));
        const int b  = (int)( gm >> 18);
        const float* xb = x + (long long)b * DEPTH * HW * HW;
        float num = 0.f, den = 0.f;
        const int obeg = (lane < 16) ? 0  : 14;
        const int oend = (lane < 16) ? 14 : NOUT;
        for (int o = obeg; o < oend; ++o) {
            const float w  = sWgt[wave][m][o];
            const int oz = o / 9, r2 = o - oz * 9;
            const int oy = r2 / 3, ox = r2 - oy * 3;
            const int y2 = y  + oy - 1;        // H/W zero padded, depth valid
            const int x2 = xw + ox - 1;
            float xv = 0.f;
            if ((unsigned)y2 < (unsigned)HW && (unsigned)x2 < (unsigned)HW)
                xv = xb[((dz + oz) * HW + y2) * HW + x2];
            num += w * xv;
            den += w;
        }
        num += __shfl_xor(num, 16, 32);
        den += __shfl_xor(den, 16, 32);
        if (lane < 16)
            out[gm] = num / den;
    }
}

extern "C" void kernel_launch(void* const* d_in, const int* in_sizes, int n_in,
                              void* d_out, int out_size, void* d_ws, size_t ws_size,
                              hipStream_t stream) {
    (void)n_in; (void)d_ws; (void)ws_size; (void)out_size;
    const float* x  = (const float*)d_in[0];
    const float* dn = (const float*)d_in[1];
    const float* gi = (const float*)d_in[2];
    const float* dw = (const float*)d_in[3];
    const float* db = (const float*)d_in[4];
    const float* rw = (const float*)d_in[5];
    const float* rb = (const float*)d_in[6];
    float* out = (float*)d_out;

    const int M = in_sizes[1] / PATCH;         // 524288 voxels
    dim3 grid(M / 64), block(128);             // 4 waves/block, 16 voxels/wave
    hipLaunchKernelGGL(jbf_wmma_kernel, grid, block, 0, stream,
                       x, dn, gi, dw, db, rw, rb, out);
}